// Memory_69595650064523
// MI455X (gfx1250) — compile-verified
//
#include <hip/hip_runtime.h>
#include <hip/hip_bf16.h>

typedef __attribute__((ext_vector_type(2))) float v2f;
typedef __attribute__((ext_vector_type(4))) float v4f;
typedef __attribute__((ext_vector_type(8))) float v8f;

#define NROWS 2000
#define DDIM  25088
#define BB    8

// workspace layout (float offsets)
#define WS_SIM   0        // 8 x 2000 raw dot(key_b, mem_n)
#define WS_MEMN  16000    // 2000 sum-of-squares per memory row
#define WS_KEYN  18000    // 8 sum-of-squares per key row
#define WS_G     18008    // 64 Gram matrix
#define WS_W     18072    // 8 x 2000 final weights
#define WS_ZERO_COUNT 18008

__global__ __launch_bounds__(256) void k0_zero(float* __restrict__ ws) {
    int i = blockIdx.x * 256 + threadIdx.x;
    if (i < WS_ZERO_COUNT) ws[i] = 0.0f;
}

// one b128 per operand feeds two WMMAs: WMMA1 sums d-offsets {0,1,4,5},
// WMMA2 sums {2,3,6,7}; A and B use the same K->d map so the result is exact.
#define K1_STEP(A4, B4, DO_KNRM)                                                  \
    {                                                                             \
        v4f am_ = (A4) * msk4;                                                    \
        v2f a1_ = {am_.x, am_.y}, a2_ = {am_.z, am_.w};                           \
        v2f b1_ = {(B4).x, (B4).y}, b2_ = {(B4).z, (B4).w};                       \
        acc = __builtin_amdgcn_wmma_f32_16x16x4_f32(false, a1_, false, b1_,       \
                                                    (short)0, acc, false, false); \
        acc = __builtin_amdgcn_wmma_f32_16x16x4_f32(false, a2_, false, b2_,       \
                                                    (short)0, acc, false, false); \
        nrm += (B4).x*(B4).x + (B4).y*(B4).y + (B4).z*(B4).z + (B4).w*(B4).w;     \
        if (DO_KNRM)                                                              \
            knrm += am_.x*am_.x + am_.y*am_.y + am_.z*am_.z + am_.w*am_.w;        \
    }

// Pass 1: sim[b][n] = dot(key[b], mem[n]); also ||mem[n]||^2 (+ ||key[b]||^2 in tile 0).
// grid = (8 d-slices, 125 n-tiles), 8 waves/block, each wave owns 392 d-values.
__global__ __launch_bounds__(256) void k1_sim(const float* __restrict__ mem,
                                              const float* __restrict__ key,
                                              float* __restrict__ ws) {
    __shared__ float lds_sim[128];
    __shared__ float lds_norm[16];
    __shared__ float lds_keyn[8];

    const int tid   = threadIdx.x;
    const int wave  = tid >> 5;
    const int lane  = tid & 31;
    const int slice = blockIdx.x;           // 0..7
    const int tile  = blockIdx.y;           // 0..124
    const int n0    = tile * 16;
    const int m     = lane & 15;            // A row (batch) / B column (mem row in tile)
    const int hoff  = (lane < 16) ? 0 : 4;  // lane's half of each 8-wide d-window

    if (tid < 128) lds_sim[tid]  = 0.0f;
    if (tid < 16)  lds_norm[tid] = 0.0f;
    if (tid < 8)   lds_keyn[tid] = 0.0f;
    __syncthreads();

    const int dbase = slice * 3136 + wave * 392;     // 8*8*392 = 25088 exact
    const float* pm = mem + (size_t)(n0 + m) * DDIM + dbase + hoff;
    const float* pk = key + (size_t)(m & 7) * DDIM + dbase + hoff;  // valid row; masked
    const float msk = (m < BB) ? 1.0f : 0.0f;
    const v4f msk4 = {msk, msk, msk, msk};

    v8f acc = {};
    float nrm  = 0.0f;
    float knrm = 0.0f;

    // software-pipelined: issue next iteration's loads before consuming current regs
    v4f b4 = *(const v4f*)pm;
    v4f a4 = *(const v4f*)pk;
    if (tile == 0) {
        for (int it = 0; it < 48; ++it) {
            const v4f b4n = *(const v4f*)(pm + 8);
            const v4f a4n = *(const v4f*)(pk + 8);
            __builtin_prefetch(pm + 72, 0, 3);
            K1_STEP(a4, b4, true);
            a4 = a4n; b4 = b4n; pm += 8; pk += 8;
        }
        K1_STEP(a4, b4, true);
    } else {
        for (int it = 0; it < 48; ++it) {
            const v4f b4n = *(const v4f*)(pm + 8);
            const v4f a4n = *(const v4f*)(pk + 8);
            __builtin_prefetch(pm + 72, 0, 3);
            K1_STEP(a4, b4, false);
            a4 = a4n; b4 = b4n; pm += 8; pk += 8;
        }
        K1_STEP(a4, b4, false);
    }

    // C layout: VGPR i, lanes 0-15 -> (M=i, N=lane). Rows 8..15 are zero (A padded).
#pragma unroll
    for (int i = 0; i < 8; ++i)
        if (lane < 16) atomicAdd(&lds_sim[i * 16 + lane], acc[i]);
    atomicAdd(&lds_norm[m], nrm);
    if (tile == 0 && m < BB) atomicAdd(&lds_keyn[m], knrm);
    __syncthreads();

    if (tid < 128) {
        const int i = tid >> 4, n = tid & 15;
        atomicAdd(&ws[WS_SIM + i * NROWS + n0 + n], lds_sim[tid]);
    }
    if (tid < 16) atomicAdd(&ws[WS_MEMN + n0 + tid], lds_norm[tid]);
    if (tile == 0 && tid < 8) atomicAdd(&ws[WS_KEYN + tid], lds_keyn[tid]);
}

// Pass 2: cosine similarity -> softmax(strength*sim) -> ^sharpen -> renormalize,
// then Gram matrix G = w w^T. One block; wave b handles batch row b.
__global__ __launch_bounds__(256) void k2_softmax(const float* __restrict__ strength,
                                                  const float* __restrict__ sharpen,
                                                  float* __restrict__ ws) {
    const int tid  = threadIdx.x;
    const int wave = tid >> 5;
    const int lane = tid & 31;

    if (wave < BB) {
        const int b = wave;
        const float s  = strength[b];
        const float sh = sharpen[b];
        const float kn = fmaxf(sqrtf(ws[WS_KEYN + b]), 1e-8f);
        const float* simr = ws + WS_SIM + b * NROWS;
        const float* memn = ws + WS_MEMN;
        float* wrow = ws + WS_W + b * NROWS;

        float mx = -3.4e38f;
        for (int n = lane; n < NROWS; n += 32) {
            const float mn = fmaxf(sqrtf(memn[n]), 1e-8f);
            const float z  = s * (simr[n] / (kn * mn));
            wrow[n] = z;
            mx = fmaxf(mx, z);
        }
        for (int o = 16; o; o >>= 1) mx = fmaxf(mx, __shfl_xor(mx, o, 32));

        float sum = 0.0f;
        for (int n = lane; n < NROWS; n += 32) {
            const float e = expf(wrow[n] - mx);
            wrow[n] = e;
            sum += e;
        }
        for (int o = 16; o; o >>= 1) sum += __shfl_xor(sum, o, 32);

        const float inv = 1.0f / sum;
        float s2 = 0.0f;
        for (int n = lane; n < NROWS; n += 32) {
            const float v = powf(wrow[n] * inv, sh);
            wrow[n] = v;
            s2 += v;
        }
        for (int o = 16; o; o >>= 1) s2 += __shfl_xor(s2, o, 32);

        const float invs = 1.0f / (s2 + 1e-16f);
        for (int n = lane; n < NROWS; n += 32) wrow[n] *= invs;
    }
    __syncthreads();

    if (tid < 64) {
        const int b = tid >> 3, b2 = tid & 7;
        const float* wa = ws + WS_W + b  * NROWS;
        const float* wb = ws + WS_W + b2 * NROWS;
        float g = 0.0f;
        for (int n = 0; n < NROWS; ++n) g += wa[n] * wb[n];
        ws[WS_G + tid] = g;
    }
}

// Pass 3: r = w @ mem + G @ add. grid = 196 blocks x 8 waves; each wave owns a
// 16-wide d-tile, reduces over 2000 rows, 8 rows (2 WMMAs) per iteration.
// B operand rows are IOFFSET-addressed (row strides fit the signed 24-bit imm).
__global__ __launch_bounds__(256) void k3_read(const float* __restrict__ mem,
                                               const float* __restrict__ add,
                                               const float* __restrict__ ws,
                                               float* __restrict__ out) {
    __shared__ float gsh[64];
    const int tid = threadIdx.x;
    if (tid < 64) gsh[tid] = ws[WS_G + tid];
    __syncthreads();

    const int wave = tid >> 5;
    const int lane = tid & 31;
    const int d0   = (blockIdx.x * 8 + wave) * 16;    // 196*8*16 = 25088 exact
    const int m    = lane & 15;
    const int hoff = (lane < 16) ? 0 : 4;             // lane's half of each 8-row window
    const float* pw = ws + WS_W + (size_t)(m & 7) * NROWS + hoff;
    const float msk = (m < BB) ? 1.0f : 0.0f;
    const v4f msk4 = {msk, msk, msk, msk};
    const float* pb = mem + (size_t)hoff * DDIM + d0 + m;   // this lane's column

    v8f acc = {};
    v4f a4 = *(const v4f*)pw;
    v2f b1, b2;
    b1.x = pb[0];
    b1.y = pb[DDIM];
    b2.x = pb[2 * DDIM];
    b2.y = pb[3 * DDIM];
    for (int it = 0; it < 249; ++it) {                // 250 8-row steps total
        const float* pbn = pb + (size_t)8 * DDIM;
        const v4f a4n = *(const v4f*)(pw + 8);
        v2f c1, c2;
        c1.x = pbn[0];
        c1.y = pbn[DDIM];
        c2.x = pbn[2 * DDIM];
        c2.y = pbn[3 * DDIM];
        __builtin_prefetch(pb + (size_t)16 * DDIM, 0, 3);
        const v4f am = a4 * msk4;
        const v2f a1 = {am.x, am.y}, a2 = {am.z, am.w};
        acc = __builtin_amdgcn_wmma_f32_16x16x4_f32(false, a1, false, b1, (short)0, acc, false, false);
        acc = __builtin_amdgcn_wmma_f32_16x16x4_f32(false, a2, false, b2, (short)0, acc, false, false);
        a4 = a4n; b1 = c1; b2 = c2; pb = pbn; pw += 8;
    }
    {
        const v4f am = a4 * msk4;
        const v2f a1 = {am.x, am.y}, a2 = {am.z, am.w};
        acc = __builtin_amdgcn_wmma_f32_16x16x4_f32(false, a1, false, b1, (short)0, acc, false, false);
        acc = __builtin_amdgcn_wmma_f32_16x16x4_f32(false, a2, false, b2, (short)0, acc, false, false);
    }

    if (lane < 16) {
        const int dd = d0 + lane;
        float av[8];
#pragma unroll
        for (int bp = 0; bp < 8; ++bp) av[bp] = add[(size_t)bp * DDIM + dd];
#pragma unroll
        for (int i = 0; i < 8; ++i) {
            float r = acc[i];
#pragma unroll
            for (int bp = 0; bp < 8; ++bp) r += gsh[i * 8 + bp] * av[bp];
            out[(size_t)i * DDIM + dd] = r;
        }
    }
}

extern "C" void kernel_launch(void* const* d_in, const int* in_sizes, int n_in,
                              void* d_out, int out_size, void* d_ws, size_t ws_size,
                              hipStream_t stream) {
    const float* mem      = (const float*)d_in[0];
    const float* key      = (const float*)d_in[1];
    const float* strength = (const float*)d_in[2];
    const float* sharpen  = (const float*)d_in[3];
    const float* add      = (const float*)d_in[4];
    float* out = (float*)d_out;
    float* ws  = (float*)d_ws;

    k0_zero<<<(WS_ZERO_COUNT + 255) / 256, 256, 0, stream>>>(ws);
    dim3 g1(8, 125);
    k1_sim<<<g1, 256, 0, stream>>>(mem, key, ws);
    k2_softmax<<<1, 256, 0, stream>>>(strength, sharpen, ws);
    k3_read<<<196, 256, 0, stream>>>(mem, add, ws, out);
}